// GEC_22814866276592
// MI455X (gfx1250) — compile-verified
//
#include <hip/hip_runtime.h>
#include <hip/hip_bf16.h>

typedef __attribute__((ext_vector_type(2))) float v2f;
typedef __attribute__((ext_vector_type(8))) float v8f;

#define N_NODES   50000
#define N_EDGES   800000
#define N_GRAPHS  50
#define IN_FEATS  128
#define H1_FEATS  64
#define OUT_FEATS 32
#define GAT_SLOPE   0.2f
#define LRELU_SLOPE 0.01f

// ---------- monotonic float<->uint key for atomic segment-max ----------
__device__ __forceinline__ unsigned flipf(float f) {
  unsigned u = __float_as_uint(f);
  return u ^ ((u >> 31) ? 0xFFFFFFFFu : 0x80000000u);
}
__device__ __forceinline__ float unflipf(unsigned u) {
  u ^= ((u >> 31) ? 0x80000000u : 0xFFFFFFFFu);
  return __uint_as_float(u);
}

// ---------- fp32 WMMA GEMM: Z(n x NO) = X(n x K) @ W(K x NO) ----------
// one wave per 16-row tile, W staged in LDS, K swept 4 at a time with
// V_WMMA_F32_16X16X4_F32 (exact fp32, matches reference precision).
template<int K, int NO>
__global__ __launch_bounds__(256) void gat_gemm(const float* __restrict__ X,
                                                const float* __restrict__ W,
                                                float* __restrict__ Z,
                                                int n_tiles)
{
  __shared__ float sW[K * NO];
  // cooperative LDS fill of the weight panel (float4, fully coalesced)
  for (int i = threadIdx.x; i < (K * NO) / 4; i += 256)
    ((float4*)sW)[i] = ((const float4*)W)[i];
  __syncthreads();

  const int tile = blockIdx.x * 8 + (threadIdx.x >> 5);
  if (tile >= n_tiles) return;                  // wave-uniform: EXEC stays all-ones
  const int lane = threadIdx.x & 31;            // wave32
  const int r    = lane & 15;                   // row within tile (A) / col (B,C)
  const int kh   = (lane >> 4) << 1;            // lanes 16..31 carry K+2/K+3

  const float* xrow = X + (size_t)(tile * 16 + r) * K;
  __builtin_prefetch(xrow, 0, 0);               // global_prefetch_b8

  v8f acc[NO / 16] = {};
  #pragma unroll 4
  for (int k0 = 0; k0 < K; k0 += 4) {
    // A 16x4 fp32 fragment: VGPR0 = K+kh, VGPR1 = K+kh+1 (ISA 7.12.2 layout)
    v2f a; a.x = xrow[k0 + kh]; a.y = xrow[k0 + kh + 1];
    #pragma unroll
    for (int t = 0; t < NO / 16; ++t) {
      // B 4x16 fragment: row K striped across lanes within a VGPR
      v2f b;
      b.x = sW[(k0 + kh)     * NO + t * 16 + r];
      b.y = sW[(k0 + kh + 1) * NO + t * 16 + r];
      acc[t] = __builtin_amdgcn_wmma_f32_16x16x4_f32(
          /*neg_a=*/false, a, /*neg_b=*/false, b,
          /*c_mod=*/(short)0, acc[t], /*reuse_a=*/false, /*reuse_b=*/false);
    }
  }

  // C/D layout: lane<16 -> rows 0..7, lane>=16 -> rows 8..15; col = lane&15
  const int rbase = (lane >> 4) * 8;
  float* zb = Z + (size_t)(tile * 16 + rbase) * NO;
  #pragma unroll
  for (int t = 0; t < NO / 16; ++t)
    #pragma unroll
    for (int i2 = 0; i2 < 8; ++i2)
      zb[(size_t)i2 * NO + t * 16 + r] = acc[t][i2];
}

// ---------- per-node attention logits: el = z.al, er = z.ar ----------
template<int NO>
__global__ void attn_logits(const float* __restrict__ Z,
                            const float* __restrict__ al,
                            const float* __restrict__ ar,
                            float* __restrict__ el, float* __restrict__ er, int n)
{
  int i = blockIdx.x * blockDim.x + threadIdx.x;
  if (i >= n) return;
  const float* zr = Z + (size_t)i * NO;
  float sl = 0.f, sr = 0.f;
  #pragma unroll 8
  for (int j = 0; j < NO; ++j) { float v = zr[j]; sl += v * al[j]; sr += v * ar[j]; }
  el[i] = sl; er[i] = sr;
}

// ---------- zero/neutral init for one layer's softmax + accumulator ----------
__global__ void init_layer(unsigned* __restrict__ mkey, float* __restrict__ ssum,
                           float* __restrict__ hacc, int n, int ntot)
{
  int i = blockIdx.x * blockDim.x + threadIdx.x;
  if (i < n) { mkey[i] = 0x007FFFFFu; /* flip(-inf) */ ssum[i] = 0.0f; }
  if (i < ntot) hacc[i] = 0.0f;
}

// ---------- edge logits + segment-max (atomic on monotonic key) ----------
__global__ void edge_logits(const int* __restrict__ src, const int* __restrict__ dst,
                            const float* __restrict__ el, const float* __restrict__ er,
                            float* __restrict__ e, unsigned* __restrict__ mkey)
{
  int i = blockIdx.x * blockDim.x + threadIdx.x;
  if (i >= N_EDGES) return;
  float v = el[src[i]] + er[dst[i]];
  v = (v > 0.f) ? v : GAT_SLOPE * v;
  e[i] = v;
  atomicMax(&mkey[dst[i]], flipf(v));
}

// ---------- p = exp(e - max), segment-sum ----------
__global__ void edge_exp(const int* __restrict__ dst, const float* __restrict__ e,
                         const unsigned* __restrict__ mkey,
                         float* __restrict__ p, float* __restrict__ ssum)
{
  int i = blockIdx.x * blockDim.x + threadIdx.x;
  if (i >= N_EDGES) return;
  int d = dst[i];
  float pv = expf(e[i] - unflipf(mkey[d]));
  p[i] = pv;
  atomicAdd(&ssum[d], pv);
}

// ---------- out[dst] += alpha * z[src] ; NO/4 threads per edge, float4 gathers ----------
template<int NO>
__global__ void edge_aggregate(const int* __restrict__ src, const int* __restrict__ dst,
                               const float* __restrict__ p, const float* __restrict__ ssum,
                               const float* __restrict__ Z, float* __restrict__ out)
{
  int gid = blockIdx.x * blockDim.x + threadIdx.x;
  constexpr int LPE = NO / 4;
  if (gid >= N_EDGES * LPE) return;
  int eidx = gid / LPE;
  int f4   = (gid - eidx * LPE) * 4;
  int d = dst[eidx];
  float alpha = p[eidx] / ssum[d];
  const float4 zv = *(const float4*)(Z + (size_t)src[eidx] * NO + f4);
  float* op = out + (size_t)d * NO + f4;
  atomicAdd(op + 0, alpha * zv.x);
  atomicAdd(op + 1, alpha * zv.y);
  atomicAdd(op + 2, alpha * zv.z);
  atomicAdd(op + 3, alpha * zv.w);
}

// ---------- h = leaky_relu(h + bias, 0.01) ----------
template<int NO>
__global__ void bias_act(float* __restrict__ h, const float* __restrict__ bias, int ntot)
{
  int i = blockIdx.x * blockDim.x + threadIdx.x;
  if (i >= ntot) return;
  float v = h[i] + bias[i % NO];
  h[i] = (v > 0.f) ? v : LRELU_SLOPE * v;
}

// ---------- graph mean pool ----------
__global__ void pool_init(float* __restrict__ gsum, float* __restrict__ gcnt)
{
  int i = blockIdx.x * blockDim.x + threadIdx.x;
  if (i < N_GRAPHS * OUT_FEATS) gsum[i] = 0.0f;
  if (i < N_GRAPHS) gcnt[i] = 0.0f;
}
__global__ void pool_sum(const float* __restrict__ h, const int* __restrict__ gid,
                         float* __restrict__ gsum, float* __restrict__ gcnt)
{
  int i = blockIdx.x * blockDim.x + threadIdx.x;
  if (i >= N_NODES * OUT_FEATS) return;
  int node = i / OUT_FEATS, f = i - node * OUT_FEATS;
  int g = gid[node];
  atomicAdd(&gsum[g * OUT_FEATS + f], h[i]);
  if (f == 0) atomicAdd(&gcnt[g], 1.0f);
}
__global__ void pool_final(const float* __restrict__ gsum, const float* __restrict__ gcnt,
                           float* __restrict__ out)
{
  int i = blockIdx.x * blockDim.x + threadIdx.x;
  if (i >= N_GRAPHS * OUT_FEATS) return;
  out[i] = gsum[i] / fmaxf(gcnt[i / OUT_FEATS], 1.0f);
}

extern "C" void kernel_launch(void* const* d_in, const int* in_sizes, int n_in,
                              void* d_out, int out_size, void* d_ws, size_t ws_size,
                              hipStream_t stream)
{
  (void)in_sizes; (void)n_in; (void)out_size; (void)ws_size;
  const float* x   = (const float*)d_in[0];
  const int*   src = (const int*)d_in[1];
  const int*   dst = (const int*)d_in[2];
  const int*   gid = (const int*)d_in[3];
  const float* W1  = (const float*)d_in[4];
  const float* al1 = (const float*)d_in[5];
  const float* ar1 = (const float*)d_in[6];
  const float* b1  = (const float*)d_in[7];
  const float* W2  = (const float*)d_in[8];
  const float* al2 = (const float*)d_in[9];
  const float* ar2 = (const float*)d_in[10];
  const float* b2  = (const float*)d_in[11];
  float* out = (float*)d_out;

  // workspace carve-up (fp32 slots)
  float* ws = (float*)d_ws;
  size_t off = 0;
  float*    z1   = ws + off; off += (size_t)N_NODES * H1_FEATS;
  float*    h1   = ws + off; off += (size_t)N_NODES * H1_FEATS;
  float*    z2   = ws + off; off += (size_t)N_NODES * OUT_FEATS;
  float*    h2   = ws + off; off += (size_t)N_NODES * OUT_FEATS;
  float*    el   = ws + off; off += N_NODES;
  float*    er   = ws + off; off += N_NODES;
  float*    ebuf = ws + off; off += N_EDGES;
  float*    pbuf = ws + off; off += N_EDGES;
  unsigned* mkey = (unsigned*)(ws + off); off += N_NODES;
  float*    ssum = ws + off; off += N_NODES;
  float*    gsum = ws + off; off += (size_t)N_GRAPHS * OUT_FEATS;
  float*    gcnt = ws + off; off += N_GRAPHS;

  const int TPB = 256;
  const int tiles  = N_NODES / 16;            // 3125, exact
  const int gemm_b = (tiles + 7) / 8;         // 8 waves (wave32) per block
  const int nodeB  = (N_NODES + TPB - 1) / TPB;
  const int edgeB  = (N_EDGES + TPB - 1) / TPB;

  // ---- layer 1 ----
  gat_gemm<IN_FEATS, H1_FEATS><<<gemm_b, TPB, 0, stream>>>(x, W1, z1, tiles);
  attn_logits<H1_FEATS><<<nodeB, TPB, 0, stream>>>(z1, al1, ar1, el, er, N_NODES);
  init_layer<<<(N_NODES * H1_FEATS + TPB - 1) / TPB, TPB, 0, stream>>>(
      mkey, ssum, h1, N_NODES, N_NODES * H1_FEATS);
  edge_logits<<<edgeB, TPB, 0, stream>>>(src, dst, el, er, ebuf, mkey);
  edge_exp<<<edgeB, TPB, 0, stream>>>(dst, ebuf, mkey, pbuf, ssum);
  edge_aggregate<H1_FEATS><<<(N_EDGES * (H1_FEATS / 4) + TPB - 1) / TPB, TPB, 0, stream>>>(
      src, dst, pbuf, ssum, z1, h1);
  bias_act<H1_FEATS><<<(N_NODES * H1_FEATS + TPB - 1) / TPB, TPB, 0, stream>>>(
      h1, b1, N_NODES * H1_FEATS);

  // ---- layer 2 ----
  gat_gemm<H1_FEATS, OUT_FEATS><<<gemm_b, TPB, 0, stream>>>(h1, W2, z2, tiles);
  attn_logits<OUT_FEATS><<<nodeB, TPB, 0, stream>>>(z2, al2, ar2, el, er, N_NODES);
  init_layer<<<(N_NODES * OUT_FEATS + TPB - 1) / TPB, TPB, 0, stream>>>(
      mkey, ssum, h2, N_NODES, N_NODES * OUT_FEATS);
  edge_logits<<<edgeB, TPB, 0, stream>>>(src, dst, el, er, ebuf, mkey);
  edge_exp<<<edgeB, TPB, 0, stream>>>(dst, ebuf, mkey, pbuf, ssum);
  edge_aggregate<OUT_FEATS><<<(N_EDGES * (OUT_FEATS / 4) + TPB - 1) / TPB, TPB, 0, stream>>>(
      src, dst, pbuf, ssum, z2, h2);
  bias_act<OUT_FEATS><<<(N_NODES * OUT_FEATS + TPB - 1) / TPB, TPB, 0, stream>>>(
      h2, b2, N_NODES * OUT_FEATS);

  // ---- mean pool per graph ----
  pool_init<<<(N_GRAPHS * OUT_FEATS + TPB - 1) / TPB, TPB, 0, stream>>>(gsum, gcnt);
  pool_sum<<<(N_NODES * OUT_FEATS + TPB - 1) / TPB, TPB, 0, stream>>>(h2, gid, gsum, gcnt);
  pool_final<<<(N_GRAPHS * OUT_FEATS + TPB - 1) / TPB, TPB, 0, stream>>>(gsum, gcnt, out);
}